// EPGSimulation_90142773608462
// MI455X (gfx1250) — compile-verified
//
#include <hip/hip_runtime.h>
#include <stdint.h>
#include <math.h>

#define NS  21
#define TPB 64
#define MAX_STEPS 256

// Per-buffer LDS staging: [Fp: TPB*42 f][Fm: TPB*42 f][Z: TPB*21 f]
#define FM_OFF_F     (TPB * 2 * NS)             // float index of Fm region (in buffer)
#define Z_OFF_F      (TPB * 4 * NS)             // float index of Z region (in buffer)
#define BUF_F        (TPB * 5 * NS)             // 6720 floats = 26880 bytes per buffer
#define BUF_BYTES    (BUF_F * 4)

#define FPFM_BYTES   (TPB * NS * 16)            // 21504 B, contiguous Fp||Fm in buffer
#define FP_XFERS     (TPB * NS / 2)             // 672 b128 transfers cover Fp
#define Z_BYTES      (TPB * NS * 4)             // 5376 B
#define Z_B128_ROUNDS 5                         // 5*TPB b128 = 5120 B
#define Z_TAIL_OFF   (Z_B128_ROUNDS * TPB * 16) // 5120

#define ASYNC_PER_STEP (NS + Z_B128_ROUNDS + 1) // 27 async ops per thread per step

static_assert(Z_BYTES - Z_TAIL_OFF == TPB * 4, "Z tail must be one b32 per thread");
static_assert(ASYNC_PER_STEP == 27, "s_wait_asynccnt immediate below assumes 27");
static_assert(2 * ASYNC_PER_STEP <= 63, "ASYNCcnt is a 6-bit counter");

__global__ __launch_bounds__(TPB) void epg_scan_kernel(
    const float* __restrict__ flip, const float* __restrict__ phases,
    const float* __restrict__ T1,   const float* __restrict__ T2,
    const float* __restrict__ B0,   const float* __restrict__ B1,
    const int*   __restrict__ TRp,
    float* __restrict__ outFp, float* __restrict__ outFm, float* __restrict__ outZ,
    int B, int nsteps)
{
    __shared__ float lds[2 * BUF_F];        // double-buffered staging
    __shared__ float sFlip[MAX_STEPS];
    __shared__ float sBeta[MAX_STEPS];

    const int t  = threadIdx.x;
    const int vb = blockIdx.x * TPB;
    const int b  = vb + t;

    // Stage the (uniform) pulse train into LDS once: removes a dependent
    // global-load latency from the top of every step.
    for (int i = t; i < nsteps; i += TPB) { sFlip[i] = flip[i]; sBeta[i] = phases[i]; }

    // Per-voxel constants
    const float TRf = (float)TRp[0];
    const float e1  = expf(-TRf / T1[b]);
    const float e2  = expf(-TRf / T2[b]);
    const float rec = 1.0f - e1;
    const float phi = 6.283185307179586f * B0[b] * TRf * 0.001f;
    float pr, pim;
    sincosf(phi, &pim, &pr);
    const float er = e2 * pr;           // E2 * exp(i*phi), folded
    const float ei = e2 * pim;
    const float b1 = B1[b];

    // Register-resident state
    float fpr[NS], fpi[NS], fmr[NS], fmi[NS], zz[NS];
#pragma unroll
    for (int k = 0; k < NS; ++k) { fpr[k]=0.f; fpi[k]=0.f; fmr[k]=0.f; fmi[k]=0.f; zz[k]=0.f; }
    zz[0] = 1.0f;

    const uint32_t ldsbase = (uint32_t)(uintptr_t)(&lds[0]);

    __syncthreads();                    // sFlip/sBeta visible to all waves

    for (int n = 0; n < nsteps; ++n) {
        const float fa   = sFlip[n];
        const float beta = sBeta[n];
        float s, c;   sincosf(0.5f * fa * b1, &s, &c);
        float sb, cb; sincosf(beta, &sb, &cb);
        const float c2 = c * c, s2 = s * s, cs = c * s, c2ms2 = c2 - s2;
        const float e2r = cb * cb - sb * sb;     // Re exp(2i*beta)
        const float e2i = 2.0f * cb * sb;        // Im exp(2i*beta)

        // Fused update + roll:
        //   new Fp[k] = X_Fp[k-1]  (carried temp, Fp[0] = 0)
        //   new Fm[k] = X_Fm[k+1]  (write X_Fm(k) into slot k-1; old consumed)
        //   new Z[k]  = X_Z[k]
        float pfr = 0.0f, pfi = 0.0f;
#pragma unroll
        for (int k = 0; k < NS; ++k) {
            const float fr = fpr[k], fi = fpi[k];
            const float mr = fmr[k], mi = fmi[k];
            const float zo = zz[k];
            const float csz = cs * zo;
            const float xfr = c2 * fr + s2 * (mr * e2r + mi * e2i) - csz * sb;
            const float xfi = c2 * fi + s2 * (mr * e2i - mi * e2r) + csz * cb;
            const float xmr = s2 * (fr * e2r - fi * e2i) + c2 * mr - csz * sb;
            const float xmi = -s2 * (fr * e2i + fi * e2r) + c2 * mi - csz * cb;
            float xz  = cs * ((fi - mi) * cb - (fr + mr) * sb) + c2ms2 * zo;
            xz *= e1;
            if (k == 0) xz += rec;
            zz[k] = xz;
            const float nfr = xfr * er - xfi * ei;
            const float nfi = xfr * ei + xfi * er;
            const float nmr = xmr * er + xmi * ei;
            const float nmi = xmi * er - xmr * ei;
            fpr[k] = pfr; fpi[k] = pfi;
            pfr = nfr;    pfi = nfi;                         // X_Fp(NS-1) discarded
            if (k > 0) { fmr[k-1] = nmr; fmi[k-1] = nmi; }   // X_Fm(0) discarded
        }
        fmr[NS-1] = 0.0f; fmi[NS-1] = 0.0f;

        // Double buffer: before staging into buf(n) = buf(n-2), only step n-2's
        // stores must be drained -> allow one step (27 ops) to stay in flight.
        // Async stores complete in order; the barrier then guarantees ALL waves'
        // step n-2 stores are done before any lane overwrites the buffer.
        asm volatile("s_wait_asynccnt 27" ::: "memory");
        __syncthreads();

        const int bufF = (n & 1) * BUF_F;
#pragma unroll
        for (int k = 0; k < NS; ++k) {
            lds[bufF + t * 2 * NS + 2 * k]                = fpr[k];
            lds[bufF + t * 2 * NS + 2 * k + 1]            = fpi[k];
            lds[bufF + FM_OFF_F + t * 2 * NS + 2 * k]     = fmr[k];
            lds[bufF + FM_OFF_F + t * 2 * NS + 2 * k + 1] = fmi[k];
            lds[bufF + Z_OFF_F + t * NS + k]              = zz[k];
        }
        __syncthreads();

        // Async DMA LDS -> global, b128, fully coalesced, non-temporal.
        const uint32_t labase = ldsbase + (uint32_t)((n & 1) * BUF_BYTES);
        const size_t row = (size_t)n * (size_t)B + (size_t)vb;
        const char* gFp = (const char*)outFp + row * (size_t)(NS * 8);
        const char* gFm = (const char*)outFm + row * (size_t)(NS * 8);
        const char* gZ  = (const char*)outZ  + row * (size_t)(NS * 4);

        // Fp||Fm region: TPB*NS b128 transfers -> NS per thread; per-lane base select.
        const char* gFmAdj = gFm - (size_t)(FPFM_BYTES / 2);
#pragma unroll
        for (int j = 0; j < NS; ++j) {
            const uint32_t i  = (uint32_t)(t + TPB * j);
            const char* base  = (i < FP_XFERS) ? gFp : gFmAdj;
            const uint64_t ga = (uint64_t)(base + (size_t)i * 16u);
            const uint32_t la = labase + i * 16u;
            asm volatile("global_store_async_from_lds_b128 %0, %1, off th:TH_STORE_NT"
                         :: "v"(ga), "v"(la) : "memory");
        }
        // Z region: 5 b128 rounds + one b32 tail per thread.
#pragma unroll
        for (int j = 0; j < Z_B128_ROUNDS; ++j) {
            const uint32_t i  = (uint32_t)(t + TPB * j);
            const uint64_t ga = (uint64_t)(gZ + (size_t)i * 16u);
            const uint32_t la = labase + (uint32_t)FPFM_BYTES + i * 16u;
            asm volatile("global_store_async_from_lds_b128 %0, %1, off th:TH_STORE_NT"
                         :: "v"(ga), "v"(la) : "memory");
        }
        {
            const uint64_t ga = (uint64_t)(gZ + (size_t)Z_TAIL_OFF + (size_t)t * 4u);
            const uint32_t la = labase + (uint32_t)(FPFM_BYTES + Z_TAIL_OFF) + (uint32_t)t * 4u;
            asm volatile("global_store_async_from_lds_b32 %0, %1, off th:TH_STORE_NT"
                         :: "v"(ga), "v"(la) : "memory");
        }
    }
    // Drain before wave end (S_ENDPGM also implies wait-idle; belt and braces).
    asm volatile("s_wait_asynccnt 0" ::: "memory");
}

extern "C" void kernel_launch(void* const* d_in, const int* in_sizes, int n_in,
                              void* d_out, int out_size, void* d_ws, size_t ws_size,
                              hipStream_t stream) {
    (void)n_in; (void)out_size; (void)d_ws; (void)ws_size;

    const float* flip   = (const float*)d_in[0];
    const float* phases = (const float*)d_in[1];
    const float* T1     = (const float*)d_in[2];
    const float* T2     = (const float*)d_in[3];
    const float* B0     = (const float*)d_in[4];
    const float* B1     = (const float*)d_in[5];
    const int*   TRp    = (const int*)d_in[6];
    // d_in[7] = TE, unused by the reference.

    const int nsteps = in_sizes[0];       // 48
    const int B      = in_sizes[2];       // 32768

    // Output: Fp_all (complex64) || Fm_all (complex64) || Z_all (float32),
    // complex stored as interleaved float pairs.
    const size_t C = (size_t)nsteps * (size_t)B * (size_t)NS;
    float* outFp = (float*)d_out;
    float* outFm = outFp + 2 * C;
    float* outZ  = outFp + 4 * C;

    const int blocks = B / TPB;           // 32768/64 = 512
    epg_scan_kernel<<<blocks, TPB, 0, stream>>>(flip, phases, T1, T2, B0, B1, TRp,
                                                outFp, outFm, outZ, B, nsteps);
}